// NNUEModel_52037823758706
// MI455X (gfx1250) — compile-verified
//
#include <hip/hip_runtime.h>

#define NUM_FEATURES 768
#define HIDDEN 1024
#define BATCH 8192
#define KFEAT 32
#define TILE_B 16
#define ROW_PAD (2 * HIDDEN + 4)   // 2052 floats: pad rotates banks by 4 per M-lane

typedef __attribute__((ext_vector_type(2))) float v2f;
typedef __attribute__((ext_vector_type(8))) float v8f;

__device__ __forceinline__ float screlu(float x) {
    x = fminf(fmaxf(x, 0.0f), 1.0f);
    return x * x;
}

__global__ __launch_bounds__(256)
void nnue_fwd_kernel(const int* __restrict__ wfeat,
                     const int* __restrict__ bfeat,
                     const int* __restrict__ stm,
                     const float* __restrict__ W,      // (768, 1024)
                     const float* __restrict__ ftb,    // (1024,)
                     const float* __restrict__ ow,     // (2048,)
                     const float* __restrict__ ob,     // (1,)
                     float* __restrict__ out)          // (8192,)
{
    extern __shared__ char smem_raw[];
    float* lds_comb = (float*)smem_raw;                 // TILE_B * ROW_PAD floats
    float* lds_ow   = lds_comb + TILE_B * ROW_PAD;      // 2048 floats
    int*   lds_idx  = (int*)(lds_ow + 2 * HIDDEN);      // 2 * TILE_B * KFEAT ints
    int*   lds_stm  = lds_idx + 2 * TILE_B * KFEAT;     // TILE_B ints

    const int tile = blockIdx.x;       // 0..511
    const int t    = threadIdx.x;      // 0..255
    const int b0   = tile * TILE_B;

    // ---- Phase 0: stage indices / stm / out_weight in LDS ----
    for (int i = t; i < TILE_B * KFEAT; i += 256) {
        lds_idx[i]                   = wfeat[b0 * KFEAT + i];
        lds_idx[TILE_B * KFEAT + i]  = bfeat[b0 * KFEAT + i];
    }
    for (int i = t; i < 2 * HIDDEN; i += 256) lds_ow[i] = ow[i];
    if (t < TILE_B) lds_stm[t] = stm[b0 + t];
    __syncthreads();

    // ---- Phase 1: gather-sum + SCReLU + stm select, per batch item ----
    const int h = t * 4;                               // each thread owns 4 hidden slots
    const float4 bias4 = *(const float4*)(ftb + h);

    for (int b = 0; b < TILE_B; ++b) {
        float4 aw = bias4;
        float4 ab = bias4;
        const int* wi = lds_idx + b * KFEAT;
        const int* bi = lds_idx + TILE_B * KFEAT + b * KFEAT;

#pragma unroll 8
        for (int k = 0; k < KFEAT; ++k) {
            const int iw = wi[k];
            if (iw >= 0) {
                const float4 r = *(const float4*)(W + (size_t)iw * HIDDEN + h);
                aw.x += r.x; aw.y += r.y; aw.z += r.z; aw.w += r.w;
            }
            const int ib = bi[k];
            if (ib >= 0) {
                const float4 r = *(const float4*)(W + (size_t)ib * HIDDEN + h);
                ab.x += r.x; ab.y += r.y; ab.z += r.z; ab.w += r.w;
            }
            // stay ahead in L2/L0: prefetch rows 8 features out (global_prefetch_b8)
            if (k + 8 < KFEAT) {
                const int pw = wi[k + 8];
                if (pw >= 0) __builtin_prefetch(W + (size_t)pw * HIDDEN + h, 0, 1);
                const int pb = bi[k + 8];
                if (pb >= 0) __builtin_prefetch(W + (size_t)pb * HIDDEN + h, 0, 1);
            }
        }

        float4 wa, ba;
        wa.x = screlu(aw.x); wa.y = screlu(aw.y); wa.z = screlu(aw.z); wa.w = screlu(aw.w);
        ba.x = screlu(ab.x); ba.y = screlu(ab.y); ba.z = screlu(ab.z); ba.w = screlu(ab.w);

        const int sm = lds_stm[b];                     // 0: us=white, 1: us=black
        const float4 us   = sm ? ba : wa;
        const float4 them = sm ? wa : ba;

        float* row = lds_comb + b * ROW_PAD;
        *(float4*)(row + h)          = us;             // combined[b][0:1024]   = us
        *(float4*)(row + HIDDEN + h) = them;           // combined[b][1024:2048]= them
    }
    __syncthreads();

    // ---- Phase 2: out[b] = combined[b] . out_weight via V_WMMA_F32_16X16X4_F32 ----
    // D[m,n] = sum_k A[m,k] * B[k,n]; A = combined rows (per-M), B = ow chunk
    // replicated across N. Accumulate C over the 2048-dim contraction in K=4 chunks.
    if (t < 32) {
        const int lane = t;
        const int m    = lane & 15;
        const int o    = (lane >> 4) ? 2 : 0;          // 32-bit frag: lanes 16-31 hold K=2,3
        const float* arow = lds_comb + m * ROW_PAD;

        v8f c0 = {}, c1 = {}, c2 = {}, c3 = {};
        for (int j = 0; j < 2 * HIDDEN; j += 16) {
            const v2f a0 = *(const v2f*)(arow + j + o);
            const v2f w0 = *(const v2f*)(lds_ow + j + o);
            c0 = __builtin_amdgcn_wmma_f32_16x16x4_f32(false, a0, false, w0, (short)0, c0, false, false);

            const v2f a1 = *(const v2f*)(arow + j + 4 + o);
            const v2f w1 = *(const v2f*)(lds_ow + j + 4 + o);
            c1 = __builtin_amdgcn_wmma_f32_16x16x4_f32(false, a1, false, w1, (short)0, c1, false, false);

            const v2f a2 = *(const v2f*)(arow + j + 8 + o);
            const v2f w2 = *(const v2f*)(lds_ow + j + 8 + o);
            c2 = __builtin_amdgcn_wmma_f32_16x16x4_f32(false, a2, false, w2, (short)0, c2, false, false);

            const v2f a3 = *(const v2f*)(arow + j + 12 + o);
            const v2f w3 = *(const v2f*)(lds_ow + j + 12 + o);
            c3 = __builtin_amdgcn_wmma_f32_16x16x4_f32(false, a3, false, w3, (short)0, c3, false, false);
        }
        const v8f c = c0 + c1 + c2 + c3;

        // C/D layout: VGPR v, lanes 0-15 -> (M=v,   N=lane),
        //                      lanes 16-31 -> (M=v+8, N=lane-16). Column N=0 has our dots.
        const float obias = *ob;
        if (lane == 0) {
#pragma unroll
            for (int v = 0; v < 8; ++v) out[b0 + v] = c[v] + obias;
        } else if (lane == 16) {
#pragma unroll
            for (int v = 0; v < 8; ++v) out[b0 + 8 + v] = c[v] + obias;
        }
    }
}

extern "C" void kernel_launch(void* const* d_in, const int* in_sizes, int n_in,
                              void* d_out, int out_size, void* d_ws, size_t ws_size,
                              hipStream_t stream) {
    const int*   wfeat = (const int*)d_in[0];
    const int*   bfeat = (const int*)d_in[1];
    const int*   stm   = (const int*)d_in[2];
    const float* W     = (const float*)d_in[3];
    const float* ftb   = (const float*)d_in[4];
    const float* ow    = (const float*)d_in[5];
    const float* ob    = (const float*)d_in[6];
    float*       out   = (float*)d_out;

    const size_t lds_bytes =
        (size_t)(TILE_B * ROW_PAD + 2 * HIDDEN) * sizeof(float) +
        (size_t)(2 * TILE_B * KFEAT + TILE_B) * sizeof(int);   // ~140 KB (< 320 KB WGP LDS)

    dim3 grid(BATCH / TILE_B);   // 512 workgroups
    dim3 block(256);             // 8 waves (wave32)
    nnue_fwd_kernel<<<grid, block, lds_bytes, stream>>>(wfeat, bfeat, stm, W, ftb, ow, ob, out);
}